// DeformConvBound_20109036880556
// MI455X (gfx1250) — compile-verified
//
#include <hip/hip_runtime.h>

// ---------------------------------------------------------------------------
// Deformable conv (DCN) for MI455X / gfx1250.
//   B=8, CIN=64, COUT=64, H=W=128, 3x3, stride=pad=dil=1, K=9, BOUND=8
// Stage 1: offset conv (64->18 ch, 3x3) -> d_ws, clipped to [-8, 8].
// Stage 2: fused bilinear gather + WMMA f32 GEMM (M=COUT, K=CIN*9, N=pos).
//   Block = 256 threads (8 waves): 64 couts x 32 positions per workgroup.
//   Wave w -> cout block (w%4), position half-tile (w/4).
// ---------------------------------------------------------------------------

#define Bn   8
#define CIN  64
#define COUT 64
#define Hs   128
#define Ws   128
#define KK   9
#define CK   (CIN * KK)        // 576
#define NOFF 18                // 2*K offset channels
#define PLANE (Hs * Ws)        // 16384
#define NPOS 32                // output positions per workgroup (2 WMMA N tiles)

typedef __attribute__((ext_vector_type(2))) float v2f;
typedef __attribute__((ext_vector_type(8))) float v8f;

// ---------------------------------------------------------------------------
// Kernel 1: 3x3 offset conv. One block per (b, ho) row, 128 threads = wo.
// x rows are staged in LDS once per input channel and reused by all 18
// output channels; off_w reads are wave-uniform -> scalar loads.
// ---------------------------------------------------------------------------
__global__ __launch_bounds__(128)
void dcn_offset_conv(const float* __restrict__ x,
                     const float* __restrict__ off_w,
                     const float* __restrict__ off_b,
                     float* __restrict__ offs) {
    const int blk = blockIdx.x;          // 0 .. B*H-1
    const int b   = blk / Hs;
    const int ho  = blk % Hs;
    const int wo  = threadIdx.x;         // 0..127

    __shared__ float rows[3][Ws + 2];    // zero-padded row window

    float acc[NOFF];
#pragma unroll
    for (int ch = 0; ch < NOFF; ++ch) acc[ch] = off_b[ch];

    for (int c = 0; c < CIN; ++c) {
        const float* xp = x + ((size_t)(b * CIN + c)) * PLANE;
#pragma unroll
        for (int r = 0; r < 3; ++r) {
            const int hy = ho - 1 + r;
            rows[r][wo + 1] = (hy >= 0 && hy < Hs) ? xp[hy * Ws + wo] : 0.0f;
        }
        if (wo == 0) {
#pragma unroll
            for (int r = 0; r < 3; ++r) { rows[r][0] = 0.0f; rows[r][Ws + 1] = 0.0f; }
        }
        __syncthreads();

        // 9 window values, reused by all 18 output channels.
        float win[9];
#pragma unroll
        for (int kh = 0; kh < 3; ++kh)
#pragma unroll
            for (int kw = 0; kw < 3; ++kw)
                win[kh * 3 + kw] = rows[kh][wo + kw];

#pragma unroll
        for (int ch = 0; ch < NOFF; ++ch) {
            const float* wp = off_w + ((size_t)ch * CIN + c) * 9;  // uniform -> SMEM
            float s = 0.0f;
#pragma unroll
            for (int t = 0; t < 9; ++t) s = fmaf(win[t], wp[t], s);
            acc[ch] += s;
        }
        __syncthreads();
    }

#pragma unroll
    for (int ch = 0; ch < NOFF; ++ch) {
        float v = fminf(fmaxf(acc[ch], -8.0f), 8.0f);
        offs[(((size_t)b * NOFF + ch) * Hs + ho) * Ws + wo] = v;
    }
}

// ---------------------------------------------------------------------------
// Kernel 2: fused bilinear gather + fp32 WMMA GEMM.
// Block = 256 threads (8 waves). Tile = 64 couts x 32 positions.
// Wave w computes D[16*(w%4) .. +15, 16*(w/4) .. +15] with
// V_WMMA_F32_16X16X4_F32 (full fp32, matches reference numerics).
// ---------------------------------------------------------------------------
__global__ __launch_bounds__(256)
void dcn_gather_wmma(const float* __restrict__ x,
                     const float* __restrict__ weight,   // [COUT][CIN*9]
                     const float* __restrict__ offs,     // [B][18][H][W]
                     float* __restrict__ out) {
    const int blk   = blockIdx.x;                 // B * H * (W/32) = 4096
    const int tiles = Ws / NPOS;                  // 4
    const int b     = blk / (Hs * tiles);
    const int rem   = blk % (Hs * tiles);
    const int ho    = rem / tiles;
    const int wo0   = (rem % tiles) * NPOS;
    const int tid   = threadIdx.x;

    __shared__ float s_acc[CK * NPOS];            // 576 x 32 fp32 panel (72 KB)
    __shared__ int   s_idx[KK * NPOS * 4];        // clamped tap indices
    __shared__ float s_wgt[KK * NPOS * 4];        // bilinear weights (validity folded)

    // ---- Stage A: per-(k, pos) bilinear parameters (288 items, 256 threads)
    for (int i = tid; i < KK * NPOS; i += 256) {
        const int k   = i / NPOS;
        const int pos = i % NPOS;
        const int wo  = wo0 + pos;
        const float oy = offs[(((size_t)b * NOFF + 2 * k    ) * Hs + ho) * Ws + wo];
        const float ox = offs[(((size_t)b * NOFF + 2 * k + 1) * Hs + ho) * Ws + wo];
        const float sy = oy + (float)(k / 3) + (float)(ho - 1);
        const float sx = ox + (float)(k % 3) + (float)(wo - 1);
        const float y0f = floorf(sy), x0f = floorf(sx);
        const float ty = sy - y0f,   tx = sx - x0f;
        const int y0 = (int)y0f, x0 = (int)x0f;
#pragma unroll
        for (int t = 0; t < 4; ++t) {
            const int dy = t >> 1, dx = t & 1;
            const int yc = y0 + dy, xc = x0 + dx;
            const bool valid = (yc >= 0) && (yc < Hs) && (xc >= 0) && (xc < Ws);
            const float wy = dy ? ty : 1.0f - ty;
            const float wx = dx ? tx : 1.0f - tx;
            const int ycc = min(max(yc, 0), Hs - 1);
            const int xcc = min(max(xc, 0), Ws - 1);
            s_idx[i * 4 + t] = ycc * Ws + xcc;
            s_wgt[i * 4 + t] = valid ? (wy * wx) : 0.0f;
        }
    }
    __syncthreads();

    // ---- Stage B: gather 4 taps x 64 channels into the LDS acc panel.
    // Threads: pos = tid%32 (fast), c8 = tid/32; each (k) iteration covers
    // 64 channels x 32 positions; tap params are broadcast within the pos group.
    {
        const int pos = tid & (NPOS - 1);
        const int c8  = tid >> 5;                 // 0..7
#pragma unroll
        for (int k = 0; k < KK; ++k) {
            const int i = k * NPOS + pos;
            const int   i0 = s_idx[i * 4 + 0], i1 = s_idx[i * 4 + 1];
            const int   i2 = s_idx[i * 4 + 2], i3 = s_idx[i * 4 + 3];
            const float w0 = s_wgt[i * 4 + 0], w1 = s_wgt[i * 4 + 1];
            const float w2 = s_wgt[i * 4 + 2], w3 = s_wgt[i * 4 + 3];
#pragma unroll
            for (int cc = 0; cc < CIN / 8; ++cc) {
                const int c = cc * 8 + c8;
                const float* plane = x + ((size_t)(b * CIN + c)) * PLANE;
                float v = fmaf(w0, plane[i0],
                          fmaf(w1, plane[i1],
                          fmaf(w2, plane[i2], w3 * plane[i3])));
                s_acc[(c * KK + k) * NPOS + pos] = v;   // ck = c*9+k matches W layout
            }
        }
    }
    __syncthreads();

    // ---- Stage C: per-wave 16x16 fp32 WMMA tile, reduction over CK=576.
    const int wave  = tid >> 5;                   // 0..7
    const int cblk  = wave & 3;                   // cout block (x16)
    const int ptile = wave >> 2;                  // position half-tile (x16)
    const int lane  = tid & 31;
    const int m     = lane & 15;                  // A row / B col / D col
    const int half  = lane >> 4;                  // K-pair select per ISA layout

    v8f d = {};
    const float* Wm = weight + ((size_t)(cblk * 16 + m)) * CK;
    const float* Bp = s_acc + ptile * 16 + m;     // column base within panel

#pragma unroll 4
    for (int ck0 = 0; ck0 < CK; ck0 += 4) {
        // A (16x4 f32): VGPR0 = K(2*half), VGPR1 = K(2*half+1), M = lane%16
        v2f a;
        a.x = Wm[ck0 + 2 * half];
        a.y = Wm[ck0 + 2 * half + 1];
        // B (4x16 f32): mirrored K striping, N = lane%16 within this half-tile
        v2f bb;
        bb.x = Bp[(ck0 + 2 * half    ) * NPOS];
        bb.y = Bp[(ck0 + 2 * half + 1) * NPOS];
        d = __builtin_amdgcn_wmma_f32_16x16x4_f32(
                /*neg_a=*/false, a, /*neg_b=*/false, bb,
                /*c_mod=*/(short)0, d, /*reuse_a=*/false, /*reuse_b=*/false);
    }

    // ---- Stage D: store D. VGPR r: lanes0-15 -> M=r, lanes16-31 -> M=r+8.
    const int n = ptile * 16 + m;                 // position within block tile
#pragma unroll
    for (int r = 0; r < 8; ++r) {
        const int cout = cblk * 16 + r + 8 * half;
        out[(((size_t)b * COUT + cout) * Hs + ho) * Ws + wo0 + n] = d[r];
    }
}

// ---------------------------------------------------------------------------
extern "C" void kernel_launch(void* const* d_in, const int* in_sizes, int n_in,
                              void* d_out, int out_size, void* d_ws, size_t ws_size,
                              hipStream_t stream) {
    const float* x      = (const float*)d_in[0];
    const float* weight = (const float*)d_in[1];
    const float* off_w  = (const float*)d_in[2];
    const float* off_b  = (const float*)d_in[3];
    float*       out    = (float*)d_out;
    float*       offs   = (float*)d_ws;          // B*18*H*W fp32 = 9.4 MB

    dcn_offset_conv<<<Bn * Hs, 128, 0, stream>>>(x, off_w, off_b, offs);

    const int nblk = Bn * Hs * (Ws / NPOS);      // 4096
    dcn_gather_wmma<<<nblk, 256, 0, stream>>>(x, weight, offs, out);
}